// mult_layer_71511205478967
// MI455X (gfx1250) — compile-verified
//
#include <hip/hip_runtime.h>

// out[b,h,w,i*8+j] = x[b,h,w,i] * weight[h,w,j] + bias
// x: (32,112,112,8) f32, weight: (112,112,8) f32, bias: (1,) f32
// out: (32,112,112,64) f32
//
// Bandwidth-bound (~116 MB @ 23.3 TB/s ~= 5us). One wave per (h,w) pixel:
// 8x V_WMMA_F32_16X16X4_F32 produce the full 32x64 output block.
// Bias is folded into the matmul (A upper-lane K=2 column = 1.0, B K=2 row =
// bias) so the C operand is the inline-0 SRC2 -> zero accumulator copies.

typedef __attribute__((ext_vector_type(2))) float v2f;
typedef __attribute__((ext_vector_type(8))) float v8f;

#define HEIGHT 112
#define WIDTH  112
#define HW     (HEIGHT * WIDTH)   // 12544
#define BATCH  32

__global__ __launch_bounds__(256) void outer_wmma_kernel(
    const float* __restrict__ x,
    const float* __restrict__ wgt,
    const float* __restrict__ bias,
    float* __restrict__ out)
{
    const int lane = threadIdx.x & 31;
    const int wave = threadIdx.x >> 5;
    const int hw   = blockIdx.x * (blockDim.x >> 5) + wave;   // wave-uniform
    if (hw >= HW) return;   // uniform branch; EXEC all-1 past here

    const float bval = bias[0];
    const bool  lo16 = lane < 16;

    // Weight row: unconditional mirrored load; all lanes hit one 32B segment.
    const float wv = wgt[hw * 8 + (lane & 7)];

    // ---- B operand (4x16), one per wave, shared by all 8 WMMAs ----
    // row K=0: w[j] in N=0..7 | row K=1: w[j] in N=8..15
    // row K=2: bias (multiplied by A's 1.0 column) | row K=3: 0
    // Layout: VGPR0 = rows K=0 (lanes 0-15) / K=2 (lanes 16-31),
    //         VGPR1 = rows K=1 / K=3  (mirrors documented A/C half-lane split)
    v2f bmat;
    bmat.x = lo16 ? ((lane < 8) ? wv : 0.0f) : bval;
    bmat.y = (lo16 && lane >= 8) ? wv : 0.0f;

    // ---- Preload all A data: 8x global_load_b64 in one clause ----
    // Mirrored addressing: lanes 16-31 reload lanes 0-15's bytes (same cache
    // line, coalesced) instead of exec-masked loads -> no saveexec/cbranch,
    // all loads in flight before the WMMA chain.
    const int m = lane & 15;
    const float* xbase = x + ((size_t)m * HW + (size_t)hw) * 8;
    v2f xa[8];
#pragma unroll
    for (int u = 0; u < 8; ++u) {
        const int bh = u >> 2;            // batch half
        const int t  = u & 3;             // channel pair (2t, 2t+1)
        xa[u] = *(const v2f*)(xbase + (size_t)bh * 16 * HW * 8 + 2 * t);
    }

    const int n   = lane & 15;            // N index within D tile
    const int mhi = (lane >> 4) << 3;     // lanes 16-31 hold M = v+8 in D
    const v8f cz  = {};                   // zero C -> inline 0 SRC2

#pragma unroll
    for (int u = 0; u < 8; ++u) {
        const int bh = u >> 2;
        const int t  = u & 3;

        // A (16x4): lanes 0-15 = x channels (2t,2t+1) for 16 batches (K=0,1);
        // lanes 16-31 = (1.0, 0.0) constants (K=2 pairs with bias row of B).
        v2f amat;
        amat.x = lo16 ? xa[u].x : 1.0f;
        amat.y = lo16 ? xa[u].y : 0.0f;

        // D[m,n] = x[b,2t]*w + x[b,2t+1]*w + 1*bias + 0
        v8f d = __builtin_amdgcn_wmma_f32_16x16x4_f32(
            /*neg_a=*/false, amat,
            /*neg_b=*/false, bmat,
            /*c_mod=*/(short)0, cz,
            /*reuse_a=*/false, /*reuse_b=*/false);

        // Store: D VGPR v -> batch b = bh*16 + v (+8 upper lanes),
        // channel c = t*16 + n; each store = 2x 64B contiguous segments.
#pragma unroll
        for (int v = 0; v < 8; ++v) {
            const int b = bh * 16 + v + mhi;
            out[((size_t)b * HW + (size_t)hw) * 64 + (size_t)(t * 16 + n)] =
                d[v];
        }
    }
}

extern "C" void kernel_launch(void* const* d_in, const int* in_sizes, int n_in,
                              void* d_out, int out_size, void* d_ws, size_t ws_size,
                              hipStream_t stream) {
    const float* x    = (const float*)d_in[0];
    const float* wgt  = (const float*)d_in[1];
    const float* bias = (const float*)d_in[2];
    float* out        = (float*)d_out;

    // 8 waves per 256-thread block, one wave per pixel: 12544 / 8 = 1568 blocks.
    dim3 block(256);
    dim3 grid(HW / 8);
    outer_wmma_kernel<<<grid, block, 0, stream>>>(x, wgt, bias, out);
}